// GNNModel_30193620090945
// MI455X (gfx1250) — compile-verified
//
#include <hip/hip_runtime.h>
#include <hip/hip_bf16.h>

#define N_NODES 100000
#define N_EDGES 800000
#define IN_DIM  256
#define HID     128
#define OUT_DIM 64

typedef float v2f __attribute__((ext_vector_type(2)));
typedef float v8f __attribute__((ext_vector_type(8)));

// ---------------------------------------------------------------------------
// Utility kernels
// ---------------------------------------------------------------------------
__global__ void zero_kernel(float* __restrict__ p, int n) {
    int i = blockIdx.x * blockDim.x + threadIdx.x;
    if (i < n) p[i] = 0.0f;
}

__global__ void degree_kernel(const int* __restrict__ dst, float* __restrict__ degs, int n_edges) {
    int e = blockIdx.x * blockDim.x + threadIdx.x;
    if (e < n_edges) atomicAdd(&degs[dst[e]], 1.0f);
}

__global__ void norm_kernel(float* __restrict__ norm, int n) {
    int i = blockIdx.x * blockDim.x + threadIdx.x;
    if (i < n) {
        float d = norm[i];
        norm[i] = rsqrtf(fmaxf(d, 1.0f));
    }
}

// ---------------------------------------------------------------------------
// Edge scatter: agg[dst] += h[src] * norm[src]   (pre-norm fused into gather)
// One wave handles one edge's 128 features: lane handles 4 consecutive floats.
// ---------------------------------------------------------------------------
__global__ void scatter_kernel(const float* __restrict__ h,
                               const float* __restrict__ norm,
                               const int* __restrict__ src,
                               const int* __restrict__ dst,
                               float* __restrict__ agg, int n_edges) {
    int gid = blockIdx.x * blockDim.x + threadIdx.x;
    int e = gid >> 5;
    if (e >= n_edges) return;
    int l = gid & 31;
    int s = src[e];
    int d = dst[e];
    float ns = norm[s];
    const float4 v = *reinterpret_cast<const float4*>(h + (size_t)s * HID + l * 4);
    float* out = agg + (size_t)d * HID + l * 4;
    atomicAdd(out + 0, v.x * ns);
    atomicAdd(out + 1, v.y * ns);
    atomicAdd(out + 2, v.z * ns);
    atomicAdd(out + 3, v.w * ns);
}

// ---------------------------------------------------------------------------
// WMMA fp32 strip GEMM: one wave computes a full 16-row x N output strip.
//   C[16,N] = op( (norm? * A[16,K]) @ W[K,N] + bias, relu? ) + resid?
// NT = N/16 accumulator tiles live in registers (NT*8 VGPRs). Each A fragment
// is loaded ONCE per k-step and replayed against all NT B tiles -> A is read
// exactly once from HBM; W stays hot in WGP$/L2.
//
// Fragment layouts per CDNA5 ISA (V_WMMA_F32_16X16X4_F32):
//   A 16x4: lanes 0-15 hold K={0,1} of row lane&15; lanes 16-31 hold K={2,3}
//   B 4x16: mirrored by column; C/D: VGPR r <-> row r + 8*(lane>>4), col lane&15
// ---------------------------------------------------------------------------
template<int NT>
__global__ __launch_bounds__(32) void wmma_gemm_strip(
    const float* __restrict__ A, const float* __restrict__ W,
    const float* __restrict__ bias, const float* __restrict__ norm,
    const float* __restrict__ resid, float* __restrict__ C,
    int M, int K, int do_relu)
{
    constexpr int N = NT * 16;
    const int lane  = threadIdx.x;        // 0..31
    const int half  = lane >> 4;          // 0 or 1
    const int idx16 = lane & 15;
    const int m0 = blockIdx.x * 16;

    const int arow = m0 + idx16;
    const int koff = half * 2;

    const float nscale = (norm != nullptr) ? norm[arow] : 1.0f;
    const float* __restrict__ Arow = A + (size_t)arow * K;

    v8f acc[NT];
#pragma unroll
    for (int t = 0; t < NT; ++t) acc[t] = (v8f){};

    for (int k = 0; k < K; k += 4) {
        // A fragment: 2 consecutive floats, 8B-aligned -> single b64 load
        v2f a = *reinterpret_cast<const v2f*>(Arow + k + koff);
        a.x *= nscale;
        a.y *= nscale;
        const float* __restrict__ Wr0 = W + (size_t)(k + koff) * N + idx16;
        const float* __restrict__ Wr1 = Wr0 + N;
#pragma unroll
        for (int t = 0; t < NT; ++t) {
            v2f b;
            b.x = Wr0[t * 16];
            b.y = Wr1[t * 16];
            acc[t] = __builtin_amdgcn_wmma_f32_16x16x4_f32(
                /*neg_a=*/false, a, /*neg_b=*/false, b,
                /*c_mod=*/(short)0, acc[t], /*reuse_a=*/false, /*reuse_b=*/false);
        }
    }

#pragma unroll
    for (int t = 0; t < NT; ++t) {
        const int bcol = t * 16 + idx16;
        const float bv = bias[bcol];
#pragma unroll
        for (int r = 0; r < 8; ++r) {
            int orow = m0 + half * 8 + r;
            size_t oidx = (size_t)orow * N + bcol;
            float v = acc[t][r] + bv;
            if (do_relu) v = fmaxf(v, 0.0f);
            if (resid != nullptr) v += resid[oidx];
            C[oidx] = v;
        }
    }
}

// ---------------------------------------------------------------------------
// Launch
// ---------------------------------------------------------------------------
extern "C" void kernel_launch(void* const* d_in, const int* in_sizes, int n_in,
                              void* d_out, int out_size, void* d_ws, size_t ws_size,
                              hipStream_t stream) {
    const float* features = (const float*)d_in[0];
    const int*   src      = (const int*)  d_in[1];
    const int*   dst      = (const int*)  d_in[2];
    const float* W_in     = (const float*)d_in[3];
    const float* b_in     = (const float*)d_in[4];
    const float* gcn_w0   = (const float*)d_in[5];
    const float* gcn_b0   = (const float*)d_in[6];
    const float* gcn_w1   = (const float*)d_in[7];
    const float* gcn_b1   = (const float*)d_in[8];
    const float* W_out    = (const float*)d_in[9];
    const float* b_out    = (const float*)d_in[10];
    float* out = (float*)d_out;

    // Workspace carve-up (256B aligned slabs)
    char* ws = (char*)d_ws;
    float* norm = (float*)ws;                             // N_NODES floats
    ws += ((size_t)N_NODES * sizeof(float) + 255) & ~(size_t)255;
    float* H = (float*)ws;                                // N_NODES*HID floats
    ws += ((size_t)N_NODES * HID * sizeof(float) + 255) & ~(size_t)255;
    float* AGG = (float*)ws;                              // N_NODES*HID floats

    const int nodeVec = N_NODES * HID;                    // 12.8M
    const int mBlocks = N_NODES / 16;                     // 6250

    // 1) degrees -> norm = clip(deg,1)^-0.5   (computed in-place in `norm`)
    zero_kernel<<<(N_NODES + 255) / 256, 256, 0, stream>>>(norm, N_NODES);
    degree_kernel<<<(N_EDGES + 255) / 256, 256, 0, stream>>>(dst, norm, N_EDGES);
    norm_kernel<<<(N_NODES + 255) / 256, 256, 0, stream>>>(norm, N_NODES);

    // 2) H = features @ W_in + b_in          (K=256, N=128 -> NT=8)
    wmma_gemm_strip<8><<<mBlocks, 32, 0, stream>>>(
        features, W_in, b_in, nullptr, nullptr, H, N_NODES, IN_DIM, /*relu=*/0);

    // 3) Layer 0: AGG = scatter(norm*H); H = H + relu(norm*AGG @ w0 + b0)
    zero_kernel<<<(nodeVec + 255) / 256, 256, 0, stream>>>(AGG, nodeVec);
    scatter_kernel<<<(N_EDGES * 32 + 255) / 256, 256, 0, stream>>>(
        H, norm, src, dst, AGG, N_EDGES);
    wmma_gemm_strip<8><<<mBlocks, 32, 0, stream>>>(
        AGG, gcn_w0, gcn_b0, norm, /*resid=*/H, H, N_NODES, HID, /*relu=*/1);

    // 4) Layer 1 (no residual): H = relu(norm*scatter(norm*H) @ w1 + b1)
    zero_kernel<<<(nodeVec + 255) / 256, 256, 0, stream>>>(AGG, nodeVec);
    scatter_kernel<<<(N_EDGES * 32 + 255) / 256, 256, 0, stream>>>(
        H, norm, src, dst, AGG, N_EDGES);
    wmma_gemm_strip<8><<<mBlocks, 32, 0, stream>>>(
        AGG, gcn_w1, gcn_b1, norm, nullptr, H, N_NODES, HID, /*relu=*/1);

    // 5) out = H @ W_out + b_out             (K=128, N=64 -> NT=4)
    wmma_gemm_strip<4><<<mBlocks, 32, 0, stream>>>(
        H, W_out, b_out, nullptr, nullptr, out, N_NODES, HID, /*relu=*/0);
}